// BatchVoxelization_82884278879143
// MI455X (gfx1250) — compile-verified
//
#include <hip/hip_runtime.h>
#include <stdint.h>

// ---------------- problem constants (from the reference) ----------------
#define B_      4
#define N_      200000
#define C_      4
#define GX_     512
#define GY_     512
#define GZ_     1
#define NVOX    (GX_ * GY_ * GZ_)      // 262144
#define MAXV    20000
#define MAXP    30
#define TILE    1024
#define NTILES  ((N_ + TILE - 1) / TILE)   // 196
#define NPAD    (NTILES * TILE)            // 200704
#define HSZ     4096                        // per-tile LDS hash (load factor <= 25%)

// ---------------- TDM descriptor vector types (CDNA5) ----------------
typedef unsigned int tdm_v4u __attribute__((ext_vector_type(4)));
typedef int          tdm_v8i __attribute__((ext_vector_type(8)));
typedef int          tdm_v4i __attribute__((ext_vector_type(4)));

// ============================================================
// Kernel 0: initialize outputs + per-batch voxel count tables.
// Reference semantics: voxels=0, coors rows = [b,-1,-1,-1], counts=0.
// ============================================================
__global__ void k_init(float* __restrict__ vox, long long* __restrict__ coors,
                       int* __restrict__ npts, int* __restrict__ voxcount) {
    size_t i = (size_t)blockIdx.x * blockDim.x + threadIdx.x;
    size_t stride = (size_t)gridDim.x * blockDim.x;

    const size_t NVF = (size_t)B_ * MAXV * MAXP * C_;   // 9.6M floats
    for (size_t k = i; k < NVF; k += stride) vox[k] = 0.0f;

    const size_t NCO = (size_t)B_ * MAXV * 4;           // 320K int64
    for (size_t k = i; k < NCO; k += stride) {
        int col = (int)(k & 3);
        long long b = (long long)(k / ((size_t)4 * MAXV));
        coors[k] = (col == 0) ? b : -1ll;
    }

    for (size_t k = i; k < (size_t)B_ * MAXV; k += stride) npts[k] = 0;
    for (size_t k = i; k < (size_t)B_ * NVOX; k += stride) voxcount[k] = 0;
}

// ============================================================
// Kernel 1: per-point linear voxel id (-1 if invalid / padding)
// + per-voxel point count via atomics (sum is order-independent
// => bit-deterministic).
// ============================================================
__global__ void k_lin(const float* __restrict__ pts, int* __restrict__ lin,
                      int* __restrict__ voxcount) {
    int g = blockIdx.x * blockDim.x + threadIdx.x;
    int b = g / NPAD;
    int i = g - b * NPAD;
    if (b >= B_) return;
    if (i >= N_) { lin[g] = -1; return; }

    const float* p = pts + ((size_t)b * N_ + i) * C_;
    float x = p[0], y = p[1], z = p[2];
    int cx = (int)floorf((x + 51.2f) / 0.2f);
    int cy = (int)floorf((y + 51.2f) / 0.2f);
    int cz = (int)floorf((z + 5.0f) / 8.0f);
    bool valid = (cx >= 0) & (cx < GX_) & (cy >= 0) & (cy < GY_) &
                 (cz >= 0) & (cz < GZ_);
    if (valid) {
        int l = (cz * GY_ + cy) * GX_ + cx;
        lin[g] = l;
        atomicAdd(&voxcount[(size_t)b * NVOX + l], 1);
    } else {
        lin[g] = -1;
    }
}

// ============================================================
// Kernel 2: one 1024-thread WG per batch. Exclusive scan of voxel
// presence over the 262144-entry table => gid in ascending-lin
// order (the reference's stable-sort group numbering). Emits coors
// + min(count,30), builds gid table, zeroes voxcount for reuse as
// the cross-tile running offsets.
// ============================================================
__global__ void __launch_bounds__(1024)
k_assign(int* __restrict__ voxcount, int* __restrict__ gidtab,
         long long* __restrict__ coors, int* __restrict__ npts) {
    __shared__ int part[1024];
    int b = blockIdx.x;
    int t = threadIdx.x;
    int* vc = voxcount + (size_t)b * NVOX;
    int* gt = gidtab + (size_t)b * NVOX;
    const int CH = NVOX / 1024;  // 256
    int base0 = t * CH;

    int s = 0;
    for (int k = 0; k < CH; ++k) s += (vc[base0 + k] > 0);
    part[t] = s;
    __syncthreads();

    for (int off = 1; off < 1024; off <<= 1) {   // Kogge-Stone inclusive scan
        int v = part[t];
        int a = (t >= off) ? part[t - off] : 0;
        __syncthreads();
        part[t] = v + a;
        __syncthreads();
    }
    int run = (t > 0) ? part[t - 1] : 0;

    for (int k = 0; k < CH; ++k) {
        int v = base0 + k;
        int cnt = vc[v];
        int g = -1;
        if (cnt > 0) {
            if (run < MAXV) {
                g = run;
                int cx = v & (GX_ - 1);
                int cy = (v >> 9) & (GY_ - 1);
                int cz = v >> 18;
                size_t row = (size_t)b * MAXV + g;
                coors[row * 4 + 0] = b;
                coors[row * 4 + 1] = cz;
                coors[row * 4 + 2] = cy;
                coors[row * 4 + 3] = cx;
                npts[row] = (cnt < MAXP) ? cnt : MAXP;
            }
            run++;
        }
        gt[v] = g;
        vc[v] = 0;  // reused as running offset by k_scatter
    }
}

// ============================================================
// Kernel 3: one WG per batch; 1024-point tiles in original index
// order. Stable within-voxel rank via wave32 hierarchy:
//   lane rank  : 32-step shfl match (original order within wave)
//   wave prefix: LDS hash (CAS insert) updated in 32 wave-ordered,
//                barrier-separated phases => exact stable prefix
//   tile->tile : hash "owner" bumps the global running offset
// Tile lin[] is DMA'd into LDS by the Tensor Data Mover.
// ============================================================
__global__ void __launch_bounds__(1024)
k_scatter(const float* __restrict__ pts, const int* __restrict__ lin,
          const int* __restrict__ gidtab, int* __restrict__ runoff,
          float* __restrict__ voxout) {
    __shared__ int lin_sh[TILE];
    __shared__ int hkey[HSZ];
    __shared__ int hval[HSZ];

    int b = blockIdx.x;
    int t = threadIdx.x;
    int lane = t & 31;
    int wid = t >> 5;
    unsigned lanes_below = (lane == 0) ? 0u : (0xFFFFFFFFu >> (32 - lane));

    const int* linb = lin + (size_t)b * NPAD;
    const int* gt = gidtab + (size_t)b * NVOX;
    int* ro = runoff + (size_t)b * NVOX;
    const float* pb = pts + (size_t)b * N_ * C_;
    float* vb = voxout + (size_t)b * MAXV * MAXP * C_;

    for (int tile = 0; tile < NTILES; ++tile) {
        int i = tile * TILE + t;

        // clear per-tile hash (4 slots/thread)
        for (int k = 0; k < HSZ / TILE; ++k) {
            hkey[t * (HSZ / TILE) + k] = -1;
            hval[t * (HSZ / TILE) + k] = 0;
        }

        // prefetch next tile's point payload (global_prefetch_b8)
        if (tile + 1 < NTILES) {
            int ip = (tile + 1) * TILE + t;
            if (ip < N_) __builtin_prefetch(pb + (size_t)ip * C_, 0, 1);
        }

        // --- stage tile's lin[] into LDS via TDM ---
#if __has_builtin(__builtin_amdgcn_tensor_load_to_lds) && __has_builtin(__builtin_amdgcn_s_wait_tensorcnt)
        if (t == 0) {  // single wave issues the DMA (EXEC ignored by TDM)
            uint64_t ga = (uint64_t)(uintptr_t)(linb + tile * TILE);
            uint32_t la = (uint32_t)(uintptr_t)lin_sh;
            // D# group0: count=1 | lds_addr | global_addr[56:0] | type=2
            tdm_v4u g0 = { 1u, la, (unsigned int)ga,
                           (unsigned int)(ga >> 32) | 0x80000000u };
            // D# group1: data_size=4B; tensor_dim0=1024, tensor_dim1=1;
            // tile_dim0=1024, tile_dim1=1; dim0_stride=1024. wg_mask=0.
            tdm_v8i g1 = { (int)(2u << 16), (int)(1024u << 16), (int)(1u << 16),
                           (int)(1024u << 16), 1, 1024, 0, 0 };
            tdm_v4i g2 = { 0, 0, 0, 0 };
            tdm_v4i g3 = { 0, 0, 0, 0 };
#if defined(__clang_major__) && (__clang_major__ >= 23)
            tdm_v8i g4 = { 0, 0, 0, 0, 0, 0, 0, 0 };
            __builtin_amdgcn_tensor_load_to_lds(g0, g1, g2, g3, g4, 0);
#else
            __builtin_amdgcn_tensor_load_to_lds(g0, g1, g2, g3, 0);
#endif
            __builtin_amdgcn_s_wait_tensorcnt(0);
        }
#elif defined(__HIP_DEVICE_COMPILE__)
        // Device pass without the TDM builtin would silently lose the
        // async-tensor path; fail loudly (device pass only).
        #error "gfx1250 TDM builtin __builtin_amdgcn_tensor_load_to_lds not available"
#else
        lin_sh[t] = linb[i];  // host-side parse only
#endif
        __syncthreads();

        int v = lin_sh[t];
        bool valid = (v >= 0);

        // --- within-wave stable match (wave32) ---
        unsigned vmask = (unsigned)__ballot(valid);
        unsigned eq = 0u;
        for (int k = 0; k < 32; ++k) {
            int vk = __shfl(v, k, 32);
            eq |= ((vk == v) ? 1u : 0u) << k;
        }
        eq &= vmask;
        int rank_in_wave = __popc(eq & lanes_below);
        int leaderLane = __ffs(eq) - 1;          // lowest matching lane
        int waveCount = __popc(eq);
        bool isLeader = valid && (lane == leaderLane);

        int base = 0, g = -1;
        if (valid) {
            base = ro[v];   // points of this voxel in earlier tiles
            g = gt[v];      // kept-voxel slot, or -1 if dropped
        }

        // --- cross-wave stable prefix: 32 wave-ordered phases ---
        int prefix = 0;
        int myslot = -1, myold = 0;
        bool owner = false;
        for (int ph = 0; ph < 32; ++ph) {
            if (wid == ph && valid) {
                if (isLeader) {
                    int slot = (int)(((unsigned)v * 2654435761u) >> 20);  // 12 bits
                    for (;;) {
                        int k = hkey[slot];
                        if (k == v) { myold = hval[slot]; break; }
                        if (k == -1) {
                            int prev = atomicCAS(&hkey[slot], -1, v);
                            if (prev == -1) { owner = true; myold = hval[slot]; break; }
                            if (prev == v)  { myold = hval[slot]; break; }
                        }
                        slot = (slot + 1) & (HSZ - 1);
                    }
                    myslot = slot;
                }
                prefix = __shfl(myold, leaderLane, 32);      // earlier-wave count
                if (isLeader) hval[myslot] = myold + waveCount;
            }
            __syncthreads();
        }

        // --- emit ---
        if (valid && g >= 0) {
            int w = base + prefix + rank_in_wave;
            if (w < MAXP) {
                const float4 p = *(const float4*)(pb + (size_t)i * C_);
                *(float4*)(vb + (((size_t)g * MAXP) + w) * C_) = p;
            }
        }

        // all ro[] reads must land before the tile update
        asm volatile("s_wait_loadcnt 0\n\ts_wait_dscnt 0" ::: "memory");
        __syncthreads();
        if (owner) ro[v] = base + hval[myslot];  // exactly one writer per voxel
        __threadfence_block();
        __syncthreads();
    }
}

// ============================================================
// Host launcher. Outputs packed flat in return order:
//   voxels   f32 [B*MAXV, MAXP, C]   (38,400,000 B)
//   coors    i64 [B*MAXV, 4]         ( 2,560,000 B)
//   counts   i32 [B*MAXV]            (   320,000 B)
// Workspace (~11.2 MB): lin[B*NPAD] | voxcount[B*NVOX] | gidtab[B*NVOX]
// ============================================================
extern "C" void kernel_launch(void* const* d_in, const int* in_sizes, int n_in,
                              void* d_out, int out_size, void* d_ws, size_t ws_size,
                              hipStream_t stream) {
    const float* pts = (const float*)d_in[0];

    float* vox = (float*)d_out;
    long long* coors =
        (long long*)((char*)d_out + (size_t)B_ * MAXV * MAXP * C_ * sizeof(float));
    int* npts = (int*)((char*)coors + (size_t)B_ * MAXV * 4 * sizeof(long long));

    int* lin = (int*)d_ws;
    int* voxcount = lin + (size_t)B_ * NPAD;
    int* gidtab = voxcount + (size_t)B_ * NVOX;

    hipLaunchKernelGGL(k_init, dim3(2048), dim3(256), 0, stream,
                       vox, coors, npts, voxcount);
    hipLaunchKernelGGL(k_lin, dim3((B_ * NPAD) / 256), dim3(256), 0, stream,
                       pts, lin, voxcount);
    hipLaunchKernelGGL(k_assign, dim3(B_), dim3(1024), 0, stream,
                       voxcount, gidtab, coors, npts);
    hipLaunchKernelGGL(k_scatter, dim3(B_), dim3(1024), 0, stream,
                       pts, lin, gidtab, voxcount, vox);
}